// VectorQuantizerEMA_90606630076960
// MI455X (gfx1250) — compile-verified
//
#include <hip/hip_runtime.h>

// VQ-VAE nearest-codebook lookup on gfx1250.
//   z_e: (32,128,64,64) f32, embedding: (512,128) f32
//   out: (32,128,64,64) f32 = embedding[argmin_k ||z-e_k||^2]
//
// argmin ||z-e||^2 == argmax (z.e - 0.5||e||^2)  -> pure GEMM via
// v_wmma_f32_16x16x4_f32 (full f32 precision, matches reference argmin).

typedef float v2f __attribute__((ext_vector_type(2)));
typedef float v8f __attribute__((ext_vector_type(8)));

#define BATCH   32
#define CH      128
#define HW      4096          // 64*64
#define NPIX    (BATCH * HW)  // 131072
#define NCODE   512
#define KSTEPS  (CH / 4)      // 32 wmma k-steps of 4
#define CTILES  (NCODE / 16)  // 32 code tiles
#define WAVES   8             // waves per block
#define PTILES  2             // 16-pixel tiles per wave (32 px/wave)

__global__ __launch_bounds__(256) void vq_argmin_gather(
    const float* __restrict__ z,     // (B,C,H,W)
    const float* __restrict__ emb,   // (K,C)
    float* __restrict__ out)         // (B,C,H,W)
{
    __shared__ float s_norm[NCODE];      // 0.5*||e_k||^2
    __shared__ int   s_idx[WAVES][32];   // winning code per pixel slot

    const int tid  = threadIdx.x;
    const int wave = tid >> 5;
    const int lane = tid & 31;
    const int half = lane >> 4;          // 0: K 0/1, 1: K 2/3
    const int l16  = lane & 15;

    // ---- per-block: 0.5*||e_k||^2 into LDS (embedding is L2-resident) ----
    {
        const float4* r0 = (const float4*)(emb + (size_t)tid * CH);
        const float4* r1 = (const float4*)(emb + (size_t)(tid + 256) * CH);
        float s0 = 0.f, s1 = 0.f;
#pragma unroll
        for (int q = 0; q < CH / 4; ++q) {
            float4 a = r0[q], b = r1[q];
            s0 += a.x * a.x + a.y * a.y + a.z * a.z + a.w * a.w;
            s1 += b.x * b.x + b.y * b.y + b.z * b.z + b.w * b.w;
        }
        s_norm[tid]       = 0.5f * s0;
        s_norm[tid + 256] = 0.5f * s1;
    }
    __syncthreads();

    // ---- wave tile: 32 consecutive pixels (same batch image b) ----
    const int gwave   = blockIdx.x * WAVES + wave;
    const int pixBase = gwave * 32;
    const int b_img   = pixBase / HW;
    const int hwBase  = pixBase % HW;
    const float* zb   = z + (size_t)b_img * CH * HW + hwBase;

    // ---- preload A fragments: a[j][s] = z rows for 16 px, 4 channels/step ----
    // lane layout (ISA 16x4 f32 A): M = lane%16, K = 2*half + {0,1}
    v2f a[PTILES][KSTEPS];
#pragma unroll
    for (int j = 0; j < PTILES; ++j) {
        const float* zp = zb + j * 16 + l16;
#pragma unroll
        for (int s = 0; s < KSTEPS; ++s) {
            const int c0 = 4 * s + 2 * half;
            a[j][s].x = zp[(size_t)c0 * HW];
            a[j][s].y = zp[(size_t)(c0 + 1) * HW];
        }
    }

    float best[PTILES][8];
    int   bidx[PTILES][8];
#pragma unroll
    for (int j = 0; j < PTILES; ++j)
#pragma unroll
        for (int r = 0; r < 8; ++r) { best[j][r] = -3.4e38f; bidx[j][r] = 0; }

    // ---- scan code tiles: B loaded once per k-step, feeds both pixel tiles ----
    for (int t = 0; t < CTILES; ++t) {
        const int codeRow = t * 16 + l16;               // B column N = lane%16
        const float* ep = emb + (size_t)codeRow * CH + 2 * half;

        v8f acc[PTILES] = {};
#pragma unroll
        for (int s = 0; s < KSTEPS; ++s) {
            v2f bf = *(const v2f*)(ep + 4 * s);         // contiguous pair: K=2h,2h+1
#pragma unroll
            for (int j = 0; j < PTILES; ++j)
                acc[j] = __builtin_amdgcn_wmma_f32_16x16x4_f32(
                    false, a[j][s], false, bf, (short)0, acc[j], false, false);
        }

        const float sn = s_norm[codeRow];
#pragma unroll
        for (int j = 0; j < PTILES; ++j)
#pragma unroll
            for (int r = 0; r < 8; ++r) {
                const float sc = acc[j][r] - sn;        // row M = r + 8*half
                if (sc > best[j][r]) { best[j][r] = sc; bidx[j][r] = codeRow; }
            }
    }

    // ---- argmax reduce over the 16 lanes sharing each row (tie -> min idx) ----
#pragma unroll
    for (int j = 0; j < PTILES; ++j)
#pragma unroll
        for (int r = 0; r < 8; ++r) {
            float s = best[j][r];
            int   i = bidx[j][r];
#pragma unroll
            for (int m = 8; m >= 1; m >>= 1) {
                float so = __shfl_xor(s, m, 16);
                int   io = __shfl_xor(i, m, 16);
                if (so > s || (so == s && io < i)) { s = so; i = io; }
            }
            if (l16 == 0)
                s_idx[wave][j * 16 + r + 8 * half] = i;  // pixel slot = tile row
        }
    __syncthreads();

    // ---- gather winning rows, store coalesced in (B,C,H,W) ----
    const int myidx = s_idx[wave][lane];                 // pixel pixBase+lane
    const float* erow = emb + (size_t)myidx * CH;        // L2 hit
    float* ob = out + (size_t)b_img * CH * HW + hwBase + lane;
#pragma unroll 4
    for (int c = 0; c < CH; ++c)
        ob[(size_t)c * HW] = erow[c];                    // 128B line per instr
}

extern "C" void kernel_launch(void* const* d_in, const int* in_sizes, int n_in,
                              void* d_out, int out_size, void* d_ws, size_t ws_size,
                              hipStream_t stream) {
    const float* z   = (const float*)d_in[0];
    const float* emb = (const float*)d_in[1];
    float* out       = (float*)d_out;
    (void)in_sizes; (void)n_in; (void)out_size; (void)d_ws; (void)ws_size;

    const int blocks = NPIX / (WAVES * PTILES * 16);     // 512
    vq_argmin_gather<<<blocks, 256, 0, stream>>>(z, emb, out);
}